// GraphConvolutionLayer_2078764172294
// MI455X (gfx1250) — compile-verified
//
#include <hip/hip_runtime.h>
#include <hip/hip_bf16.h>
#include <math.h>

typedef __attribute__((ext_vector_type(2))) float v2f;
typedef __attribute__((ext_vector_type(8))) float v8f;

#define DFEAT 128
#define STAT_BLOCKS 512

// ---------------------------------------------------------------------------
// 1) Column statistics: per-block partial sum / sum-of-squares (deterministic)
// ---------------------------------------------------------------------------
__global__ __launch_bounds__(256) void gcn_stats_kernel(const float* __restrict__ feat,
                                                        float* __restrict__ psum,
                                                        float* __restrict__ psq,
                                                        int n) {
    __shared__ float ls[DFEAT], lq[DFEAT];
    const int c    = threadIdx.x & (DFEAT - 1);   // column 0..127
    const int half = threadIdx.x >> 7;            // 0 or 1 (two rows per step)
    float s = 0.f, q = 0.f;
    for (int r = blockIdx.x * 2 + half; r < n; r += STAT_BLOCKS * 2) {
        float v = feat[(size_t)r * DFEAT + c];
        s += v;
        q += v * v;
    }
    if (half) { ls[c] = s; lq[c] = q; }
    __syncthreads();
    if (!half) {
        psum[blockIdx.x * DFEAT + c] = s + ls[c];
        psq [blockIdx.x * DFEAT + c] = q + lq[c];
    }
}

// ---------------------------------------------------------------------------
// 2) Fold partials -> per-column affine (scale, shift) for batchnorm
// ---------------------------------------------------------------------------
__global__ __launch_bounds__(128) void gcn_reduce_kernel(const float* __restrict__ psum,
                                                         const float* __restrict__ psq,
                                                         const float* __restrict__ gamma,
                                                         const float* __restrict__ beta,
                                                         float* __restrict__ scale,
                                                         float* __restrict__ shift,
                                                         int n) {
    const int c = threadIdx.x;
    float s = 0.f, q = 0.f;
    for (int b = 0; b < STAT_BLOCKS; ++b) {
        s += psum[b * DFEAT + c];
        q += psq [b * DFEAT + c];
    }
    const float inv_n = 1.0f / (float)n;
    const float mean  = s * inv_n;
    const float var   = q * inv_n - mean * mean;
    const float inv   = rsqrtf(var + 1e-5f);
    const float sc    = gamma[c] * inv;
    scale[c] = sc;
    shift[c] = beta[c] - mean * sc;
}

// ---------------------------------------------------------------------------
// 3) Normalize: h = feat * scale + shift  (float4, coalesced)
// ---------------------------------------------------------------------------
__global__ __launch_bounds__(256) void gcn_normalize_kernel(const float* __restrict__ feat,
                                                            const float* __restrict__ scale,
                                                            const float* __restrict__ shift,
                                                            float* __restrict__ h,
                                                            size_t total4) {
    size_t idx = (size_t)blockIdx.x * blockDim.x + threadIdx.x;   // float4 index
    if (idx >= total4) return;
    const int cg = (int)(idx & 31);                               // 32 float4 groups per row
    float4 f  = ((const float4*)feat)[idx];
    float4 sc = ((const float4*)scale)[cg];
    float4 sh = ((const float4*)shift)[cg];
    float4 r;
    r.x = f.x * sc.x + sh.x;
    r.y = f.y * sc.y + sh.y;
    r.z = f.z * sc.z + sh.z;
    r.w = f.w * sc.w + sh.w;
    ((float4*)h)[idx] = r;
}

// ---------------------------------------------------------------------------
// 4) Edge scatter: one wave (32 lanes) per edge, float4 per lane.
//    agg[dst] += h[src] ; deg[dst] += 1
// ---------------------------------------------------------------------------
__global__ __launch_bounds__(256) void gcn_scatter_kernel(const float* __restrict__ h,
                                                          const int* __restrict__ src,
                                                          const int* __restrict__ dst,
                                                          float* __restrict__ agg,
                                                          float* __restrict__ deg,
                                                          int n_edges) {
    long long gid   = (long long)blockIdx.x * blockDim.x + threadIdx.x;
    long long total = (long long)n_edges * 32;
    long long step  = (long long)gridDim.x * blockDim.x;
    for (; gid < total; gid += step) {
        const int e    = (int)(gid >> 5);
        const int lane = (int)(gid & 31);
        const int s = src[e];
        const int d = dst[e];
        float4 v = *(const float4*)(h + (size_t)s * DFEAT + lane * 4);
        float* ap = agg + (size_t)d * DFEAT + lane * 4;
        atomicAdd(ap + 0, v.x);
        atomicAdd(ap + 1, v.y);
        atomicAdd(ap + 2, v.z);
        atomicAdd(ap + 3, v.w);
        if (lane == 0) atomicAdd(deg + d, 1.0f);
    }
}

// ---------------------------------------------------------------------------
// 5) WMMA GEMM epilogue kernel.
//    Block = 16 rows x 128 cols of output. 8 waves; wave w owns 16x16 tile
//    at cols [16w, 16w+16). C = h*Wself + (agg/deg)*Wneigh over K=128 in
//    steps of 4 via V_WMMA_F32_16X16X4_F32, then + b + h (skip) -> exact GELU.
//
//    A lane map (ISA 7.12.2): M = lane&15, K = 2*(lane>>4)+v  -> float2 load.
//    C lane map:              N = lane&15, M = r + 8*(lane>>4).
// ---------------------------------------------------------------------------
__global__ __launch_bounds__(256) void gcn_wmma_gemm_kernel(const float* __restrict__ h,
                                                            const float* __restrict__ agg,
                                                            const float* __restrict__ deg,
                                                            const float* __restrict__ Wself,
                                                            const float* __restrict__ Wneigh,
                                                            const float* __restrict__ bias,
                                                            float* __restrict__ out,
                                                            int n) {
    const int wave  = threadIdx.x >> 5;           // 0..7 -> N tile
    const int lane  = threadIdx.x & 31;
    const int m     = lane & 15;
    const int khalf = lane >> 4;                  // 0/1
    const int row_base = blockIdx.x * 16;
    const int row   = row_base + m;               // A-operand row for this lane
    const int ncol  = wave * 16 + m;              // B-operand / C column for this lane

    const float rdeg = 1.0f / fmaxf(deg[row], 1.0f);

    v8f c = {};

    // --- h @ W_self ---
    #pragma unroll 4
    for (int k0 = 0; k0 < DFEAT; k0 += 4) {
        const int ka = k0 + 2 * khalf;
        v2f a = *(const v2f*)(h + (size_t)row * DFEAT + ka);
        v2f b;
        b.x = Wself[(size_t)ka * DFEAT + ncol];
        b.y = Wself[(size_t)(ka + 1) * DFEAT + ncol];
        c = __builtin_amdgcn_wmma_f32_16x16x4_f32(false, a, false, b,
                                                  (short)0, c, false, false);
    }

    // --- (agg/deg) @ W_neigh  (degree scale folded into A load) ---
    #pragma unroll 4
    for (int k0 = 0; k0 < DFEAT; k0 += 4) {
        const int ka = k0 + 2 * khalf;
        v2f a = *(const v2f*)(agg + (size_t)row * DFEAT + ka);
        a.x *= rdeg;
        a.y *= rdeg;
        v2f b;
        b.x = Wneigh[(size_t)ka * DFEAT + ncol];
        b.y = Wneigh[(size_t)(ka + 1) * DFEAT + ncol];
        c = __builtin_amdgcn_wmma_f32_16x16x4_f32(false, a, false, b,
                                                  (short)0, c, false, false);
    }

    // --- epilogue: + bias + skip(h) -> exact GELU ---
    const float bn = bias[ncol];
    #pragma unroll
    for (int r = 0; r < 8; ++r) {
        const int orow = row_base + r + 8 * khalf;
        if (orow < n) {
            const float hv = h[(size_t)orow * DFEAT + ncol];
            const float x  = c[r] + bn + hv;
            out[(size_t)orow * DFEAT + ncol] = 0.5f * x * (1.0f + erff(x * 0.70710678118654752f));
        }
    }
}

// ---------------------------------------------------------------------------
// Host-side launcher
// ---------------------------------------------------------------------------
extern "C" void kernel_launch(void* const* d_in, const int* in_sizes, int n_in,
                              void* d_out, int out_size, void* d_ws, size_t ws_size,
                              hipStream_t stream) {
    const float* feat   = (const float*)d_in[0];
    const int*   src    = (const int*)  d_in[1];
    const int*   dst    = (const int*)  d_in[2];
    const float* gamma  = (const float*)d_in[3];
    const float* beta   = (const float*)d_in[4];
    const float* Wself  = (const float*)d_in[5];
    const float* Wneigh = (const float*)d_in[6];
    const float* bias   = (const float*)d_in[7];
    float*       out    = (float*)d_out;

    const int n = in_sizes[0] / DFEAT;     // 100000 nodes
    const int e = in_sizes[1];             // 800000 edges

    // Workspace layout (floats)
    float* ws     = (float*)d_ws;
    float* psum   = ws;                                   // STAT_BLOCKS*128
    float* psq    = psum  + (size_t)STAT_BLOCKS * DFEAT;  // STAT_BLOCKS*128
    float* scale  = psq   + (size_t)STAT_BLOCKS * DFEAT;  // 128
    float* shift  = scale + DFEAT;                        // 128
    float* h      = shift + DFEAT;                        // n*128
    float* agg    = h     + (size_t)n * DFEAT;            // n*128
    float* deg    = agg   + (size_t)n * DFEAT;            // n

    // 1) column stats
    gcn_stats_kernel<<<STAT_BLOCKS, 256, 0, stream>>>(feat, psum, psq, n);

    // 2) fold -> scale/shift
    gcn_reduce_kernel<<<1, 128, 0, stream>>>(psum, psq, gamma, beta, scale, shift, n);

    // 3) normalize
    const size_t total4 = (size_t)n * (DFEAT / 4);
    const int nblk = (int)((total4 + 255) / 256);
    gcn_normalize_kernel<<<nblk, 256, 0, stream>>>(feat, scale, shift, h, total4);

    // zero agg + deg (contiguous region) each call — required for atomics
    hipMemsetAsync(agg, 0, ((size_t)n * DFEAT + (size_t)n) * sizeof(float), stream);

    // 4) edge scatter
    const long long sthreads = (long long)e * 32;
    int sblocks = (int)((sthreads + 255) / 256);
    if (sblocks > 262144) sblocks = 262144;   // grid-stride handles remainder
    gcn_scatter_kernel<<<sblocks, 256, 0, stream>>>(h, src, dst, agg, deg, e);

    // 5) WMMA GEMM + epilogue (16 rows per block, 8 waves of 16x16 tiles)
    const int gblocks = (n + 15) / 16;
    gcn_wmma_gemm_kernel<<<gblocks, 256, 0, stream>>>(h, agg, deg, Wself, Wneigh,
                                                      bias, out, n);
}